// RXNEncoder_32212254720527
// MI455X (gfx1250) — compile-verified
//
#include <hip/hip_runtime.h>
#include <hip/hip_bf16.h>

typedef float v2f __attribute__((ext_vector_type(2)));
typedef float v8f __attribute__((ext_vector_type(8)));

constexpr int Hdim = 256;       // hidden size
constexpr int KTOT = 512;       // concat K: 256 (Wm path) + 256 (Wt path)
constexpr int ROWS = 16;        // batch rows per workgroup
constexpr int LDS_STRIDE = 516; // 512 + 4 pad -> conflict-free A reads (4*r mod 64 distinct)
constexpr int NB = 4096;        // batch
constexpr int ND = 8;           // depth

// Build W2[k][n] = (k<256) ? Wm[n][k] : Wt[n][k-256]   (i.e. [Wm^T ; Wt^T], 512x256)
// and bias2[n] = 7*bm[n] + bt[n]
__global__ __launch_bounds__(256)
void rxn_prep_kernel(const float* __restrict__ Wm, const float* __restrict__ Wt,
                     const float* __restrict__ bm, const float* __restrict__ bt,
                     float* __restrict__ W2, float* __restrict__ bias2) {
    const int n = threadIdx.x;   // 0..255
    const int k = blockIdx.x;    // 0..511
    const float v = (k < Hdim) ? Wm[n * Hdim + k] : Wt[n * Hdim + (k - Hdim)];
    W2[k * Hdim + n] = v;
    if (k == 0) bias2[n] = 7.0f * bm[n] + bt[n];
}

// One recursion level:  h_out = relu( [s|u] @ W2 + bias2 )
//   s = sum_k mask*r_emb[lid] + iflag*h_in,  u = t_emb[tid]
__global__ __launch_bounds__(256)
void rxn_level_kernel(const int*   __restrict__ template_ids, // [B][8]
                      const int*   __restrict__ leaf_ids,     // [B][8][7]
                      const float* __restrict__ leaf_mask,    // [B][8][7]
                      const float* __restrict__ internal_flag,// [8]
                      const float* __restrict__ r_emb,        // [RV][256]
                      const float* __restrict__ t_emb,        // [TV][256]
                      const float* __restrict__ W2,           // [512][256]
                      const float* __restrict__ bias2,        // [256]
                      const float* __restrict__ h_in,         // [B][256]
                      float*       __restrict__ h_out,        // [B][256]
                      int level) {
    __shared__ float smem[ROWS * LDS_STRIDE];

    const int tid  = threadIdx.x;
    const int row  = tid >> 4;                 // 0..15 : batch row within tile
    const int j    = tid & 15;                 // 16 threads cooperate per row
    const int b    = blockIdx.x * ROWS + row;  // global batch row
    const float iflag = internal_flag[level];

    // ---------------- Phase 1: gather-reduce into LDS [16 x 512] ----------------
    const int base = b * ND + level;
    const int tmpl = template_ids[base];
    float mk[7];
    int   lk[7];
#pragma unroll
    for (int k = 0; k < 7; ++k) {
        mk[k] = leaf_mask[base * 7 + k];
        lk[k] = leaf_ids[base * 7 + k];
    }

    for (int c = j; c < 128; c += 16) {        // 128 float4 chunks per row
        if (c < 64) {                          // s-part, cols [0,256)
            const int col = c * 4;
            float4 acc;
            if (iflag != 0.0f) {
                const float4 hv = *(const float4*)(h_in + (size_t)b * Hdim + col);
                acc = make_float4(hv.x * iflag, hv.y * iflag, hv.z * iflag, hv.w * iflag);
            } else {
                acc = make_float4(0.f, 0.f, 0.f, 0.f);
            }
#pragma unroll
            for (int k = 0; k < 7; ++k) {
                const float m = mk[k];
                if (m != 0.0f) {
                    const float4 rv = *(const float4*)(r_emb + (size_t)lk[k] * Hdim + col);
                    acc.x += m * rv.x; acc.y += m * rv.y;
                    acc.z += m * rv.z; acc.w += m * rv.w;
                }
            }
            *(float4*)(smem + row * LDS_STRIDE + col) = acc;
        } else {                               // u-part, cols [256,512)
            const int col = (c - 64) * 4;
            const float4 tv = *(const float4*)(t_emb + (size_t)tmpl * Hdim + col);
            *(float4*)(smem + row * LDS_STRIDE + 256 + col) = tv;
        }
    }
    __syncthreads();

    // ---------------- Phase 2: WMMA f32 16x16x4, K = 512 ----------------
    const int lane = tid & 31;   // wave32
    const int wave = tid >> 5;   // 0..7 -> owns output cols [32w, 32w+32)
    const int half = lane >> 4;  // 0/1
    const int l15  = lane & 15;

    v8f acc0 = {0.f, 0.f, 0.f, 0.f, 0.f, 0.f, 0.f, 0.f};
    v8f acc1 = {0.f, 0.f, 0.f, 0.f, 0.f, 0.f, 0.f, 0.f};

    const int n0 = wave * 32;
    const float* wp0 = W2 + n0 + l15;         // B tile 0: cols n0..n0+15
    const float* wp1 = W2 + n0 + 16 + l15;    // B tile 1: cols n0+16..n0+31
    const float* ap  = smem + l15 * LDS_STRIDE; // A: M = lane&15

#pragma unroll 4
    for (int k0 = 0; k0 < KTOT; k0 += 4) {
        const int kk = k0 + 2 * half;          // lanes 0-15: K0,K1 ; lanes 16-31: K2,K3
        const v2f a = *(const v2f*)(ap + kk);  // 8B-aligned LDS read
        v2f b0, b1;
        b0.x = wp0[(size_t)kk * Hdim];        b0.y = wp0[(size_t)(kk + 1) * Hdim];
        b1.x = wp1[(size_t)kk * Hdim];        b1.y = wp1[(size_t)(kk + 1) * Hdim];
        acc0 = __builtin_amdgcn_wmma_f32_16x16x4_f32(false, a, false, b0,
                                                     (short)0, acc0, false, false);
        acc1 = __builtin_amdgcn_wmma_f32_16x16x4_f32(false, a, false, b1,
                                                     (short)0, acc1, false, false);
    }

    // ---------------- Epilogue: bias + relu + store ----------------
    const int colg0 = n0 + l15;
    const float bb0 = bias2[colg0];
    const float bb1 = bias2[colg0 + 16];
    const int rowbase = blockIdx.x * ROWS + half * 8; // C/D layout: VGPR v -> M = v + 8*half
#pragma unroll
    for (int v = 0; v < 8; ++v) {
        const size_t rg = (size_t)(rowbase + v) * Hdim;
        float o0 = acc0[v] + bb0; o0 = o0 > 0.f ? o0 : 0.f;
        float o1 = acc1[v] + bb1; o1 = o1 > 0.f ? o1 : 0.f;
        h_out[rg + colg0]      = o0;
        h_out[rg + colg0 + 16] = o1;
    }
}

extern "C" void kernel_launch(void* const* d_in, const int* in_sizes, int n_in,
                              void* d_out, int out_size, void* d_ws, size_t ws_size,
                              hipStream_t stream) {
    const int*   template_ids  = (const int*)d_in[0];
    const int*   leaf_ids      = (const int*)d_in[1];
    const float* leaf_mask     = (const float*)d_in[2];
    const float* internal_flag = (const float*)d_in[3];
    const float* r_emb         = (const float*)d_in[4];
    const float* t_emb         = (const float*)d_in[5];
    const float* Wm            = (const float*)d_in[6];
    const float* bm            = (const float*)d_in[7];
    const float* Wt            = (const float*)d_in[8];
    const float* bt            = (const float*)d_in[9];

    float* out = (float*)d_out;
    char*  ws  = (char*)d_ws;
    float* W2    = (float*)(ws);            // 512*256*4 = 512 KB
    float* bias2 = (float*)(ws + 524288);   // 1 KB
    float* hws   = (float*)(ws + 1048576);  // 4 MB ping buffer

    rxn_prep_kernel<<<KTOT, 256, 0, stream>>>(Wm, Wt, bm, bt, W2, bias2);

    // Process levels deepest -> root: t = 0..7 maps to level l = 7-t.
    // Ping-pong: even t -> hws, odd t -> d_out; final step (t=7, l=0) lands in d_out.
    const float* hin = hws;   // never read at t=0 (internal_flag[7]==0)
    for (int t = 0; t < ND; ++t) {
        const int l = ND - 1 - t;
        float* hout = (t & 1) ? out : hws;
        rxn_level_kernel<<<NB / ROWS, 256, 0, stream>>>(
            template_ids, leaf_ids, leaf_mask, internal_flag,
            r_emb, t_emb, W2, bias2, hin, hout, l);
        hin = hout;
    }
}